// Model_39135742001279
// MI455X (gfx1250) — compile-verified
//
#include <hip/hip_runtime.h>

#define HID   2048
#define VOCAB 50257
#define NB_B  393   // ceil(50257/128)
#define NB_D  64

typedef __attribute__((ext_vector_type(2))) float v2f;
typedef __attribute__((ext_vector_type(8))) float v8f;

// ---------------------------------------------------------------------------
// Kernel 1: GRU cell. One wave per output element j (2048 waves total).
// Streams 6 weight rows (96 MB total across grid) with coalesced b128 loads.
// ---------------------------------------------------------------------------
__global__ __launch_bounds__(256) void gru_cell_kernel(
    const int*   __restrict__ tok,
    const float* __restrict__ hidden,
    const float* __restrict__ emb,
    const float* __restrict__ w_ih,
    const float* __restrict__ w_hh,
    const float* __restrict__ b_ih,
    const float* __restrict__ b_hh,
    float*       __restrict__ h_new)
{
    const int lane = threadIdx.x & 31;
    const int wave = threadIdx.x >> 5;
    const int j    = blockIdx.x * 8 + wave;        // [0, 2048)

    const float* __restrict__ x   = emb + (long)tok[0] * HID;
    const float* __restrict__ wi0 = w_ih + (long)j * HID;
    const float* __restrict__ wi1 = w_ih + (long)(j +     HID) * HID;
    const float* __restrict__ wi2 = w_ih + (long)(j + 2 * HID) * HID;
    const float* __restrict__ wh0 = w_hh + (long)j * HID;
    const float* __restrict__ wh1 = w_hh + (long)(j +     HID) * HID;
    const float* __restrict__ wh2 = w_hh + (long)(j + 2 * HID) * HID;

    float s0 = 0.f, s1 = 0.f, s2 = 0.f, s3 = 0.f, s4 = 0.f, s5 = 0.f;
    for (int k = lane * 4; k < HID; k += 128) {
        float4 xv = *(const float4*)(x + k);
        float4 hv = *(const float4*)(hidden + k);
        float4 a  = *(const float4*)(wi0 + k);
        float4 b  = *(const float4*)(wi1 + k);
        float4 c  = *(const float4*)(wi2 + k);
        float4 d  = *(const float4*)(wh0 + k);
        float4 e  = *(const float4*)(wh1 + k);
        float4 f  = *(const float4*)(wh2 + k);
        s0 += a.x * xv.x + a.y * xv.y + a.z * xv.z + a.w * xv.w;
        s1 += b.x * xv.x + b.y * xv.y + b.z * xv.z + b.w * xv.w;
        s2 += c.x * xv.x + c.y * xv.y + c.z * xv.z + c.w * xv.w;
        s3 += d.x * hv.x + d.y * hv.y + d.z * hv.z + d.w * hv.w;
        s4 += e.x * hv.x + e.y * hv.y + e.z * hv.z + e.w * hv.w;
        s5 += f.x * hv.x + f.y * hv.y + f.z * hv.z + f.w * hv.w;
    }
    #pragma unroll
    for (int off = 16; off > 0; off >>= 1) {
        s0 += __shfl_xor(s0, off, 32);
        s1 += __shfl_xor(s1, off, 32);
        s2 += __shfl_xor(s2, off, 32);
        s3 += __shfl_xor(s3, off, 32);
        s4 += __shfl_xor(s4, off, 32);
        s5 += __shfl_xor(s5, off, 32);
    }
    if (lane == 0) {
        float ir = s0 + b_ih[j];
        float iz = s1 + b_ih[j + HID];
        float in = s2 + b_ih[j + 2 * HID];
        float hr = s3 + b_hh[j];
        float hz = s4 + b_hh[j + HID];
        float hn = s5 + b_hh[j + 2 * HID];
        float r  = 1.f / (1.f + __expf(-(ir + hr)));
        float z  = 1.f / (1.f + __expf(-(iz + hz)));
        float n  = tanhf(in + r * hn);
        h_new[j] = (1.f - z) * n + z * hidden[j];
    }
}

// ---------------------------------------------------------------------------
// Kernel 2: logits = relu(w_out @ h_new + b_out) via V_WMMA_F32_16X16X4_F32.
// One wave owns 16 vocab rows. A = 16x4 weight stripe (lane=M, k-slot =
// vgpr + 2*(lane>=16), per ISA 32-bit A layout). B = h chunk broadcast
// across all 16 columns (from LDS), so every D column holds the GEMV result.
// ---------------------------------------------------------------------------
__global__ __launch_bounds__(256) void logits_wmma_kernel(
    const float* __restrict__ w_out,
    const float* __restrict__ b_out,
    const float* __restrict__ h_new,
    float*       __restrict__ y,
    float*       __restrict__ pmax)
{
    __shared__ float hs[HID];
    __shared__ float wmax[8];

    const int tid = threadIdx.x;
    for (int i = tid; i < HID / 4; i += 256)
        ((float4*)hs)[i] = ((const float4*)h_new)[i];
    __syncthreads();

    const int lane = tid & 31;
    const int wave = tid >> 5;
    const int row  = lane & 15;
    const int half = lane >> 4;
    const int m0   = blockIdx.x * 128 + wave * 16;

    long r = (long)(m0 + row);
    if (r > VOCAB - 1) r = VOCAB - 1;          // clamp OOB rows (results discarded)
    const float* __restrict__ wr = w_out + r * HID + 2 * half;
    const float* __restrict__ hb = hs + 2 * half;

    v8f acc = {};
    #pragma unroll 2
    for (int k = 0; k < HID; k += 16) {
        v2f a0 = *(const v2f*)(wr + k);
        v2f a1 = *(const v2f*)(wr + k + 4);
        v2f a2 = *(const v2f*)(wr + k + 8);
        v2f a3 = *(const v2f*)(wr + k + 12);
        v2f c0 = *(const v2f*)(hb + k);
        v2f c1 = *(const v2f*)(hb + k + 4);
        v2f c2 = *(const v2f*)(hb + k + 8);
        v2f c3 = *(const v2f*)(hb + k + 12);
        acc = __builtin_amdgcn_wmma_f32_16x16x4_f32(false, a0, false, c0, (short)0, acc, false, false);
        acc = __builtin_amdgcn_wmma_f32_16x16x4_f32(false, a1, false, c1, (short)0, acc, false, false);
        acc = __builtin_amdgcn_wmma_f32_16x16x4_f32(false, a2, false, c2, (short)0, acc, false, false);
        acc = __builtin_amdgcn_wmma_f32_16x16x4_f32(false, a3, false, c3, (short)0, acc, false, false);
    }

    // Lane 0 holds D[M=0..7][N=0] in acc[0..7]; lane 16 holds D[M=8..15][N=0].
    float lmax = 0.f;                           // relu => 0 is a valid identity
    if (lane == 0 || lane == 16) {
        #pragma unroll
        for (int i = 0; i < 8; ++i) {
            int m = m0 + half * 8 + i;
            if (m < VOCAB) {
                float v = acc[i] + b_out[m];
                v = v > 0.f ? v : 0.f;
                y[m]  = v;
                lmax  = fmaxf(lmax, v);
            }
        }
    }
    lmax = fmaxf(lmax, __shfl_xor(lmax, 16, 32));
    if (lane == 0) wmax[wave] = lmax;
    __syncthreads();
    if (tid == 0) {
        float m = wmax[0];
        #pragma unroll
        for (int i = 1; i < 8; ++i) m = fmaxf(m, wmax[i]);
        pmax[blockIdx.x] = m;
    }
}

// ---------------------------------------------------------------------------
// Kernel 3: per-block partial sums of exp(y - M). Each block deterministically
// re-reduces the 393 partial maxima first (cheap, L2-resident).
// ---------------------------------------------------------------------------
__global__ __launch_bounds__(256) void expsum_kernel(
    const float* __restrict__ y,
    const float* __restrict__ pmax,
    float*       __restrict__ psum)
{
    __shared__ float red[256];
    const int tid = threadIdx.x;

    float m = 0.f;
    for (int i = tid; i < NB_B; i += 256) m = fmaxf(m, pmax[i]);
    red[tid] = m;
    __syncthreads();
    for (int s = 128; s > 0; s >>= 1) {
        if (tid < s) red[tid] = fmaxf(red[tid], red[tid + s]);
        __syncthreads();
    }
    const float M = red[0];
    __syncthreads();

    float sum = 0.f;
    for (int i = blockIdx.x * 256 + tid; i < VOCAB; i += NB_D * 256)
        sum += __expf(y[i] - M);
    red[tid] = sum;
    __syncthreads();
    for (int s = 128; s > 0; s >>= 1) {
        if (tid < s) red[tid] += red[tid + s];
        __syncthreads();
    }
    if (tid == 0) psum[blockIdx.x] = red[0];
}

// ---------------------------------------------------------------------------
// Kernel 4: out[i] = y[i] - (M + log(sum)).
// ---------------------------------------------------------------------------
__global__ __launch_bounds__(256) void finalize_kernel(
    const float* __restrict__ y,
    const float* __restrict__ pmax,
    const float* __restrict__ psum,
    float*       __restrict__ out)
{
    __shared__ float red[256];
    const int tid = threadIdx.x;

    float m = 0.f;
    for (int i = tid; i < NB_B; i += 256) m = fmaxf(m, pmax[i]);
    red[tid] = m;
    __syncthreads();
    for (int s = 128; s > 0; s >>= 1) {
        if (tid < s) red[tid] = fmaxf(red[tid], red[tid + s]);
        __syncthreads();
    }
    const float M = red[0];
    __syncthreads();

    float ss = 0.f;
    for (int i = tid; i < NB_D; i += 256) ss += psum[i];
    red[tid] = ss;
    __syncthreads();
    for (int s = 128; s > 0; s >>= 1) {
        if (tid < s) red[tid] += red[tid + s];
        __syncthreads();
    }
    const float L = M + __logf(red[0]);

    const int i = blockIdx.x * 256 + tid;
    if (i < VOCAB) out[i] = y[i] - L;
}

// ---------------------------------------------------------------------------
extern "C" void kernel_launch(void* const* d_in, const int* in_sizes, int n_in,
                              void* d_out, int out_size, void* d_ws, size_t ws_size,
                              hipStream_t stream)
{
    const int*   tok    = (const int*)  d_in[0];
    const float* hidden = (const float*)d_in[1];
    const float* emb    = (const float*)d_in[2];
    const float* w_ih   = (const float*)d_in[3];
    const float* w_hh   = (const float*)d_in[4];
    const float* b_ih   = (const float*)d_in[5];
    const float* b_hh   = (const float*)d_in[6];
    const float* w_out  = (const float*)d_in[7];
    const float* b_out  = (const float*)d_in[8];

    float* out   = (float*)d_out;       // [VOCAB] log-softmax
    float* h_new = out + VOCAB;         // [HID] second tuple output
    float* y     = (float*)d_ws;        // [VOCAB] relu(logits), padded
    float* pmax  = y + 50432;           // [NB_B] partial maxima
    float* psum  = pmax + 512;          // [NB_D] partial exp sums

    gru_cell_kernel<<<HID / 8, 256, 0, stream>>>(tok, hidden, emb, w_ih, w_hh,
                                                 b_ih, b_hh, h_new);
    logits_wmma_kernel<<<NB_B, 256, 0, stream>>>(w_out, b_out, h_new, y, pmax);
    expsum_kernel<<<NB_D, 256, 0, stream>>>(y, pmax, psum);
    finalize_kernel<<<(VOCAB + 255) / 256, 256, 0, stream>>>(y, pmax, psum, out);
}